// bgridSlicing4DTo3D_19164144075644
// MI455X (gfx1250) — compile-verified
//
#include <hip/hip_runtime.h>
#include <stdint.h>

// ---------------------------------------------------------------------------
// Bilateral-grid slicing (4D grid -> 3D volume), MI455X / gfx1250.
//   bg : (1, 4, 32, 32, 32, 16) f32   [c, x, y, z, up], up contiguous
//   gm : (1, 1, 128, 128, 128) f32    guidance in [0,1)
//   out: (1, 4, 128, 128, 128) f32
// Memory-bound (~50 MB HBM -> ~2.1 us floor). Strategy: per-workgroup async
// staging of the tiny bg sub-grid into LDS (CDNA5 GLOBAL_LOAD_ASYNC_TO_LDS),
// then per-thread quadri-linear gather from LDS.
// ---------------------------------------------------------------------------

#define GS   32            // bg spatial size per dim
#define UP   16            // bg "up" (guidance) dim
#define NC   4             // channels
#define GO   128           // output size per dim
#define TH   4             // tile size along gh (x)
#define TW   8             // tile size along gw (y)
#define TD   8             // tile size along gd (z)
#define SX   3             // staged spatial span along x  (ceil(3*31/127)+2)
#define SY   4             // staged spatial span along y  (ceil(7*31/127)+2)
#define SZ   4             // staged spatial span along z
#define ZSTR 68            // (NC*UP)=64 floats + 4 pad -> de-stripe LDS banks
#define YSTR (SZ * ZSTR)   // 272
#define XSTR (SY * YSTR)   // 1088
#define LDSN (SX * XSTR)   // 3264 floats = 13056 B

#define AS1 __attribute__((address_space(1)))
#define AS3 __attribute__((address_space(3)))

typedef int v4i __attribute__((ext_vector_type(4)));

__device__ __forceinline__ int imin(int a, int b) { return a < b ? a : b; }

__global__ __launch_bounds__(256) void bgrid_slice_kernel(
    const float* __restrict__ bg, const float* __restrict__ gm,
    float* __restrict__ out) {
  __shared__ __align__(16) float tile[LDSN];

  const float scale = 31.0f / 127.0f;  // (h-1)/(gh-1), identical for x,y,z

  const int gz0 = blockIdx.x * TD;  // gd (fastest output dim)
  const int gy0 = blockIdx.y * TW;  // gw
  const int gx0 = blockIdx.z * TH;  // gh

  // Workgroup-uniform staged-subgrid bases (monotone fp => per-thread floor >= base).
  const int xb = (int)floorf((float)gx0 * scale);
  const int yb = (int)floorf((float)gy0 * scale);
  const int zb = (int)floorf((float)gz0 * scale);

  const int tid = threadIdx.x;

  // ---- Stage bg sub-grid into LDS: SX*SY*SZ*NC*UP = 3072 floats = 768 x b128
#pragma unroll
  for (int i = 0; i < 3; ++i) {
    const int id = tid + i * 256;        // 0..767
    const int u4 = id & 3;               // which 4-float chunk of the up dim
    const int c  = (id >> 2) & 3;
    const int z  = (id >> 4) & 3;
    const int y  = (id >> 6) & 3;
    const int x  = id >> 8;              // 0..2
    const int gx = imin(xb + x, GS - 1);
    const int gy = imin(yb + y, GS - 1);
    const int gz = imin(zb + z, GS - 1);
    const size_t goff =
        ((((size_t)c * GS + gx) * GS + gy) * GS + gz) * UP + (size_t)(u4 * 4);
    const int loff = x * XSTR + y * YSTR + z * ZSTR + c * UP + u4 * 4;
#if __has_builtin(__builtin_amdgcn_global_load_async_to_lds_b128)
    __builtin_amdgcn_global_load_async_to_lds_b128(
        (AS1 v4i*)(bg + goff), (AS3 v4i*)(&tile[loff]), 0, 0);
#else
    {
      const float* gp = bg + goff;
      unsigned lo = (unsigned)(uintptr_t)(AS3 float*)(&tile[loff]);
      asm volatile("global_load_async_to_lds_b128 %0, %1, off"
                   :
                   : "v"(lo), "v"(gp)
                   : "memory");
    }
#endif
  }

  // Guidance value for this thread's voxel (overlaps with async staging).
  const int lz = tid & 7;
  const int ly = (tid >> 3) & 7;
  const int lx = tid >> 6;  // 0..3
  const int gd = gz0 + lz;
  const int gw = gy0 + ly;
  const int gh = gx0 + lx;
  const float guide = gm[((size_t)gh * GO + gw) * GO + gd];

#if __has_builtin(__builtin_amdgcn_s_wait_asynccnt)
  __builtin_amdgcn_s_wait_asynccnt(0);
#else
  asm volatile("s_wait_asynccnt 0" ::: "memory");
#endif
  __syncthreads();

  // ---- Per-axis interpolation setup (bit-matches reference fp sequence) ----
  float cx = fminf(fmaxf((float)gh * scale, 0.0f), 31.0f);
  float cy = fminf(fmaxf((float)gw * scale, 0.0f), 31.0f);
  float cz = fminf(fmaxf((float)gd * scale, 0.0f), 31.0f);
  float fxf = floorf(cx), fyf = floorf(cy), fzf = floorf(cz);
  const int x0 = (int)fxf, y0 = (int)fyf, z0 = (int)fzf;
  const int x1 = imin(x0 + 1, GS - 1);
  const int y1 = imin(y0 + 1, GS - 1);
  const int z1 = imin(z0 + 1, GS - 1);
  const float fx = cx - fxf, fy = cy - fyf, fz = cz - fzf;

  float t = fminf(fmaxf(guide * 15.0f, 0.0f), 15.0f);
  float ftf = floorf(t);
  const int t0 = (int)ftf;
  const int t1 = imin(t0 + 1, UP - 1);
  const float ft = t - ftf, ft1 = 1.0f - ft;

  const int xi[2] = {(x0 - xb) * XSTR, (x1 - xb) * XSTR};
  const int yi[2] = {(y0 - yb) * YSTR, (y1 - yb) * YSTR};
  const int zi[2] = {(z0 - zb) * ZSTR, (z1 - zb) * ZSTR};
  const float wx[2] = {1.0f - fx, fx};
  const float wy[2] = {1.0f - fy, fy};
  const float wz[2] = {1.0f - fz, fz};

  float acc0 = 0.f, acc1 = 0.f, acc2 = 0.f, acc3 = 0.f;
#pragma unroll
  for (int a = 0; a < 2; ++a) {
#pragma unroll
    for (int b = 0; b < 2; ++b) {
#pragma unroll
      for (int d = 0; d < 2; ++d) {
        const int o = xi[a] + yi[b] + zi[d];
        const float w = wx[a] * wy[b] * wz[d];
        {
          float v = tile[o + 0 * UP + t0] * ft1 + tile[o + 0 * UP + t1] * ft;
          acc0 += w * v;
        }
        {
          float v = tile[o + 1 * UP + t0] * ft1 + tile[o + 1 * UP + t1] * ft;
          acc1 += w * v;
        }
        {
          float v = tile[o + 2 * UP + t0] * ft1 + tile[o + 2 * UP + t1] * ft;
          acc2 += w * v;
        }
        {
          float v = tile[o + 3 * UP + t0] * ft1 + tile[o + 3 * UP + t1] * ft;
          acc3 += w * v;
        }
      }
    }
  }

  const size_t ob = ((size_t)gh * GO + gw) * GO + gd;
  const size_t cs = (size_t)GO * GO * GO;  // channel stride = 2097152
  out[ob + 0 * cs] = acc0;
  out[ob + 1 * cs] = acc1;
  out[ob + 2 * cs] = acc2;
  out[ob + 3 * cs] = acc3;
}

extern "C" void kernel_launch(void* const* d_in, const int* in_sizes, int n_in,
                              void* d_out, int out_size, void* d_ws,
                              size_t ws_size, hipStream_t stream) {
  (void)in_sizes; (void)n_in; (void)d_ws; (void)ws_size; (void)out_size;
  const float* bg = (const float*)d_in[0];
  const float* gm = (const float*)d_in[1];
  float* out = (float*)d_out;
  dim3 grid(GO / TD, GO / TW, GO / TH);  // (16, 16, 32)
  bgrid_slice_kernel<<<grid, 256, 0, stream>>>(bg, gm, out);
}